// UpdateSuperpointsModule_7146825581118
// MI455X (gfx1250) — compile-verified
//
#include <hip/hip_runtime.h>

// ---------------------------------------------------------------------------
// UpdateSuperpointsModule for MI455X (gfx1250, wave32)
//   S0=4096 segments x P0=64 points, S1=1024 superpoints, C=128 features, K=3
//   Bandwidth-bound (~0.5 GB traffic).
//   - similarity GEMM: V_WMMA_F32_16X16X4_F32, one wave32 per segment
//   - neighbor-feature staging: TDM gather (TENSOR_LOAD_TO_LDS, gather_mode)
//   - 2x128MB output copies: TDM streaming (load-to-LDS + 2x store-from-LDS)
// ---------------------------------------------------------------------------

typedef float v2f __attribute__((ext_vector_type(2)));
typedef float v8f __attribute__((ext_vector_type(8)));
typedef float f4  __attribute__((ext_vector_type(4)));
typedef unsigned int u32x4 __attribute__((ext_vector_type(4)));
typedef unsigned int u32x8 __attribute__((ext_vector_type(8)));

#define S0c 4096
#define P0c 64
#define S1c 1024
#define Cc  128
#define Nc  (S0c * P0c)

// Build D# group0 word3: global_addr[56:32] | type=2 ("image")
__device__ __forceinline__ unsigned tdm_addr_hi(unsigned long long ga) {
    return ((unsigned)(ga >> 32) & 0x01FFFFFFu) | 0x80000000u; // type=2 << 30
}

// ---------------------------------------------------------------------------
// Kernel 1: per-segment assignment.
//   One wave32 per segment. TDM gather pulls the 3 neighbor feature rows
//   (sp_ids non-decreasing since l2l1 is sorted) into LDS, then the GEMM
//   points_feat[s] (64x128) x nb_feat^T (128x3) runs as 4 M-tiles x 32
//   K-steps of v_wmma_f32_16x16x4_f32. Columns 3..15 of B/D are never read,
//   so LDS rows 3..15 may stay uninitialized (no masking ALU needed; WMMA
//   raises no exceptions and D columns are independent dot products).
// ---------------------------------------------------------------------------
__global__ __launch_bounds__(32)
void assign_kernel(const float* __restrict__ sp_center_feat,   // [S1,C]
                   const float* __restrict__ sp_center_coord,  // [S1,3]
                   const float* __restrict__ points_feat,      // [S0,P0,C]
                   const float* __restrict__ points_coord,     // [S0,P0,3]
                   const int*   __restrict__ l2l1,             // [S0]
                   float* __restrict__ out_assign_f,           // [N] (float)
                   int*   __restrict__ out_assign_i)           // [N] (ws)
{
    const int s    = blockIdx.x;
    const int lane = threadIdx.x;
    const int nlo  = lane & 15;   // N column (candidate slot) / M row for A
    const int hi   = lane >> 4;   // lane-half selector

    __shared__ float nbF[16][Cc];    // rows 0..2 = neighbor feats; 3..15 unused
    __shared__ float sims[P0c][4];   // 64 points x 3 dots (padded)

    // neighbor superpoint ids (uniform; 3 clamped loads)
    int sp_ids[3];
#pragma unroll
    for (int j = 0; j < 3; ++j) {
        int nb = s - 1 + j;
        nb = nb < 0 ? 0 : (nb > S0c - 1 ? S0c - 1 : nb);
        sp_ids[j] = l2l1[nb];
    }

    // ---- TDM gather: 3 rows of 128 f32 from sp_center_feat -> nbF[0..2] ----
    {
        u32x8 g1;
        g1[0] = 0x00020000u;          // workgroup_mask=0, data_size=4B
        g1[1] = 128u << 16;           // tensor_dim0 = 128 (row length)
        g1[2] = 1024u << 16;          // tensor_dim1 = 1024 (rows in tensor)
        g1[3] = 128u << 16;           // tile_dim0 = 128
        g1[4] = 3u;                   // tile_dim1 = #valid gather indices
        g1[5] = 128u;                 // tensor_dim0_stride (elements)
        g1[6] = 0u;
        g1[7] = 0u;

        const unsigned long long ga = (unsigned long long)(size_t)sp_center_feat;
        u32x4 g0;
        g0[0] = 0x80000001u;          // count=1, gather_mode=1, 16-bit indices
        g0[1] = (unsigned)(size_t)&nbF[0][0];   // LDS byte offset (low 32 bits)
        g0[2] = (unsigned)ga;
        g0[3] = tdm_addr_hi(ga);

        u32x4 g2;                     // 16-bit row indices 0..2
        g2[0] = (unsigned)sp_ids[0] | ((unsigned)sp_ids[1] << 16);
        g2[1] = (unsigned)sp_ids[2];
        g2[2] = 0u;
        g2[3] = 0u;
        u32x4 g3 = {0u, 0u, 0u, 0u};

        asm volatile("tensor_load_to_lds %0, %1, %2, %3"
                     :: "s"(g0), "s"(g1), "s"(g2), "s"(g3)
                     : "memory");
        __builtin_amdgcn_s_wait_tensorcnt(0);
    }
    __syncthreads();

#pragma unroll
    for (int mt = 0; mt < 4; ++mt) {
        v8f acc = {};
        // A row for this lane: M = mt*16 + nlo
        const float* Arow = points_feat + ((size_t)s * P0c + mt * 16 + nlo) * Cc;
        for (int kk = 0; kk < 32; ++kk) {
            const int kb = kk * 4 + hi * 2;        // K pair per lane-half
            v2f a, b;
            a.x = Arow[kb];
            a.y = Arow[kb + 1];
            b.x = nbF[nlo][kb];                    // B[k][n] = nb_feat[n][k]
            b.y = nbF[nlo][kb + 1];
            acc = __builtin_amdgcn_wmma_f32_16x16x4_f32(
                /*neg_a=*/false, a, /*neg_b=*/false, b,
                /*c_mod=*/(short)0, acc, /*reuse_a=*/false, /*reuse_b=*/false);
        }
        // D layout: acc[i] holds M = mt*16 + i + 8*hi, N = nlo
        if (nlo < 3) {
#pragma unroll
            for (int i = 0; i < 8; ++i)
                sims[mt * 16 + hi * 8 + i][nlo] = acc[i];
        }
    }
    __syncthreads();

    // Post: 2 points per lane — distance sort + argmax + id mapping
#pragma unroll
    for (int pp = 0; pp < 2; ++pp) {
        const int p = lane + pp * 32;
        const float px = points_coord[((size_t)s * P0c + p) * 3 + 0];
        const float py = points_coord[((size_t)s * P0c + p) * 3 + 1];
        const float pz = points_coord[((size_t)s * P0c + p) * 3 + 2];

        float d0, d1, d2;
        {
            float dx, dy, dz;
            dx = px - sp_center_coord[sp_ids[0] * 3 + 0];
            dy = py - sp_center_coord[sp_ids[0] * 3 + 1];
            dz = pz - sp_center_coord[sp_ids[0] * 3 + 2];
            d0 = dx * dx + dy * dy + dz * dz;
            dx = px - sp_center_coord[sp_ids[1] * 3 + 0];
            dy = py - sp_center_coord[sp_ids[1] * 3 + 1];
            dz = pz - sp_center_coord[sp_ids[1] * 3 + 2];
            d1 = dx * dx + dy * dy + dz * dz;
            dx = px - sp_center_coord[sp_ids[2] * 3 + 0];
            dy = py - sp_center_coord[sp_ids[2] * 3 + 1];
            dz = pz - sp_center_coord[sp_ids[2] * 3 + 2];
            d2 = dx * dx + dy * dy + dz * dz;
        }
        // stable ascending 3-sort (strict-< swaps == jax.lax.top_k stability);
        // squared distance has the same order/ties as sqrt.
        int   i0 = 0, i1 = 1, i2 = 2;
        float v0 = d0, v1 = d1, v2 = d2;
        if (v1 < v0) { float tv = v0; v0 = v1; v1 = tv; int ti = i0; i0 = i1; i1 = ti; }
        if (v2 < v1) { float tv = v1; v1 = v2; v2 = tv; int ti = i1; i1 = i2; i2 = ti; }
        if (v1 < v0) { float tv = v0; v0 = v1; v1 = tv; int ti = i0; i0 = i1; i1 = ti; }

        // softmax is monotonic -> argmax over raw dots; strict-> keeps first max
        const float s0v = sims[p][i0];
        const float s1v = sims[p][i1];
        const float s2v = sims[p][i2];
        int   bestslot = i0;
        float bestv    = s0v;
        if (s1v > bestv) { bestv = s1v; bestslot = i1; }
        if (s2v > bestv) { bestslot = i2; }

        const int assigned = sp_ids[bestslot];
        out_assign_f[s * P0c + p] = (float)assigned;
        out_assign_i[s * P0c + p] = assigned;
    }
}

// ---------------------------------------------------------------------------
// Kernel 2: deterministic scatter-mean.
//   l2l1 is sorted, so points assigned to bin q come only from segments in
//   [lower_bound(q)-1, upper_bound(q)] — one block per bin scans that range
//   and reduces in fixed order (bitwise deterministic across graph replays).
// ---------------------------------------------------------------------------
__global__ __launch_bounds__(64)
void scatter_mean_kernel(const float* __restrict__ hilbert,    // [N,3]
                         const int*   __restrict__ assign_i,   // [N]
                         const int*   __restrict__ l2l1,       // [S0]
                         float* __restrict__ out_coord)        // [S1,3]
{
    const int q = blockIdx.x;
    const int t = threadIdx.x;

    // lower_bound / upper_bound of q in sorted l2l1
    int lo = 0, hib = S0c;
    while (lo < hib) { int mid = (lo + hib) >> 1; if (l2l1[mid] <  q) lo = mid + 1; else hib = mid; }
    const int lb = lo;
    lo = 0; hib = S0c;
    while (lo < hib) { int mid = (lo + hib) >> 1; if (l2l1[mid] <= q) lo = mid + 1; else hib = mid; }
    const int ub = lo;  // exclusive

    int s_begin = lb - 1; if (s_begin < 0) s_begin = 0;
    int s_end   = ub;     if (s_end > S0c - 1) s_end = S0c - 1;   // inclusive

    float sx = 0.f, sy = 0.f, sz = 0.f, cn = 0.f;
    const int i_begin = s_begin * P0c;
    const int i_end   = (s_end + 1) * P0c;
    for (int i = i_begin + t; i < i_end; i += 64) {
        if (assign_i[i] == q) {
            sx += hilbert[(size_t)i * 3 + 0];
            sy += hilbert[(size_t)i * 3 + 1];
            sz += hilbert[(size_t)i * 3 + 2];
            cn += 1.0f;
        }
    }

    __shared__ float red[64][4];
    red[t][0] = sx; red[t][1] = sy; red[t][2] = sz; red[t][3] = cn;
    __syncthreads();
#pragma unroll
    for (int off = 32; off > 0; off >>= 1) {
        if (t < off) {
            red[t][0] += red[t + off][0];
            red[t][1] += red[t + off][1];
            red[t][2] += red[t + off][2];
            red[t][3] += red[t + off][3];
        }
        __syncthreads();
    }
    if (t == 0) {
        const float c   = red[0][3];
        const float inv = 1.0f / fmaxf(c, 1.0f);
        out_coord[q * 3 + 0] = red[0][0] * inv;
        out_coord[q * 3 + 1] = red[0][1] * inv;
        out_coord[q * 3 + 2] = red[0][2] * inv;
    }
}

// ---------------------------------------------------------------------------
// Kernel 3: TDM streaming copy — rawPoint_feat read once, written twice.
//   Each block stages one 32KB tile in LDS via TENSOR_LOAD_TO_LDS, then
//   issues two TENSOR_STORE_FROM_LDS. The DMA engine moves the data; no
//   VGPR round-trip. S_ENDPGM drains TENSORcnt implicitly.
// ---------------------------------------------------------------------------
#define TILE_ELEMS 8192u            // 32 KB of f32
__global__ __launch_bounds__(32)
void tdm_copy_dual_kernel(const float* __restrict__ src,
                          float* __restrict__ d1,
                          float* __restrict__ d2)
{
    __shared__ float tile[TILE_ELEMS];
    const unsigned lds_off = (unsigned)(size_t)&tile[0];
    const unsigned long long byte_off =
        (unsigned long long)blockIdx.x * (TILE_ELEMS * 4ull);

    u32x8 g1;                         // shared 1-D tensor descriptor
    g1[0] = 0x00020000u;              // data_size = 4B
    g1[1] = (TILE_ELEMS & 0xFFFFu) << 16;  // tensor_dim0 = 8192
    g1[2] = 1u << 16;                 // tensor_dim1 = 1
    g1[3] = (TILE_ELEMS & 0xFFFFu) << 16;  // tile_dim0 = 8192
    g1[4] = 0u;                       // tile_dim1 unused
    g1[5] = TILE_ELEMS;               // tensor_dim0_stride
    g1[6] = 0u;
    g1[7] = 0u;

    {   // load tile -> LDS
        const unsigned long long ga = (unsigned long long)(size_t)src + byte_off;
        u32x4 g0;
        g0[0] = 1u;                   // count=1
        g0[1] = lds_off;
        g0[2] = (unsigned)ga;
        g0[3] = tdm_addr_hi(ga);
        asm volatile("tensor_load_to_lds %0, %1" :: "s"(g0), "s"(g1) : "memory");
    }
    __builtin_amdgcn_s_wait_tensorcnt(0);

    {   // store LDS -> points_feat_out
        const unsigned long long ga = (unsigned long long)(size_t)d1 + byte_off;
        u32x4 g0;
        g0[0] = 1u;
        g0[1] = lds_off;
        g0[2] = (unsigned)ga;
        g0[3] = tdm_addr_hi(ga);
        asm volatile("tensor_store_from_lds %0, %1" :: "s"(g0), "s"(g1) : "memory");
    }
    {   // store LDS -> hilbert_feat_level1
        const unsigned long long ga = (unsigned long long)(size_t)d2 + byte_off;
        u32x4 g0;
        g0[0] = 1u;
        g0[1] = lds_off;
        g0[2] = (unsigned)ga;
        g0[3] = tdm_addr_hi(ga);
        asm volatile("tensor_store_from_lds %0, %1" :: "s"(g0), "s"(g1) : "memory");
    }
}

// ---------------------------------------------------------------------------
// Kernel 4: sp_center_feat passthrough (B128 NT; only 512 KB).
// ---------------------------------------------------------------------------
__global__ __launch_bounds__(256)
void copy_one_kernel(const f4* __restrict__ src, f4* __restrict__ dst, int n4)
{
    const int i = blockIdx.x * 256 + threadIdx.x;
    if (i < n4) {
        const f4 v = __builtin_nontemporal_load(src + i);
        __builtin_nontemporal_store(v, dst + i);
    }
}

extern "C" void kernel_launch(void* const* d_in, const int* in_sizes, int n_in,
                              void* d_out, int out_size, void* d_ws, size_t ws_size,
                              hipStream_t stream)
{
    const float* sp_center_feat  = (const float*)d_in[0];  // [S1,C]
    const float* sp_center_coord = (const float*)d_in[1];  // [S1,3]
    const float* rawPoint_feat   = (const float*)d_in[2];  // [N,C]
    const float* hilbert_coord   = (const float*)d_in[3];  // [N,3]
    const float* points_feat     = (const float*)d_in[4];  // [S0,P0,C]
    const float* points_coord    = (const float*)d_in[5];  // [S0,P0,3]
    const int*   l2l1            = (const int*)d_in[6];    // [S0]
    (void)in_sizes; (void)n_in; (void)out_size; (void)ws_size;

    float* out = (float*)d_out;
    float* out_assign = out;                                  // N
    float* out_spfeat = out_assign + Nc;                      // S1*C
    float* out_coord  = out_spfeat + (size_t)S1c * Cc;        // S1*3
    float* out_pfeat  = out_coord  + (size_t)S1c * 3;         // N*C
    float* out_hfeat  = out_pfeat  + (size_t)Nc * Cc;         // N*C

    int* assign_i = (int*)d_ws;                               // N ints

    // 1) assignment (TDM gather + WMMA similarity + distance sort + argmax)
    assign_kernel<<<dim3(S0c), dim3(32), 0, stream>>>(
        sp_center_feat, sp_center_coord, points_feat, points_coord,
        l2l1, out_assign, assign_i);

    // 2) deterministic scatter-mean over sorted-index ranges
    scatter_mean_kernel<<<dim3(S1c), dim3(64), 0, stream>>>(
        hilbert_coord, assign_i, l2l1, out_coord);

    // 3) sp_center_feat passthrough
    {
        const int n4 = (S1c * Cc) / 4;                        // 32768
        copy_one_kernel<<<dim3((n4 + 255) / 256), dim3(256), 0, stream>>>(
            (const f4*)sp_center_feat, (f4*)out_spfeat, n4);
    }

    // 4) rawPoint_feat -> points_feat_out AND hilbert_feat_level1 via TDM
    {
        const unsigned nblocks =
            (unsigned)(((size_t)Nc * Cc) / TILE_ELEMS);       // 4096 tiles
        tdm_copy_dual_kernel<<<dim3(nblocks), dim3(32), 0, stream>>>(
            rawPoint_feat, out_pfeat, out_hfeat);
    }
}